// GIN_87729001988240
// MI455X (gfx1250) — compile-verified
//
#include <hip/hip_runtime.h>

#define N_NODES 50000
#define DIM     128
#define COUT    4

typedef __attribute__((ext_vector_type(2))) float v2f;
typedef __attribute__((ext_vector_type(4))) float v4f;
typedef __attribute__((ext_vector_type(8))) float v8f;
typedef __attribute__((ext_vector_type(4))) int   v4i;

// ---- optional CDNA5 async global->LDS copy path (guarded; safe fallback) ----
#if defined(__has_builtin)
#  if __has_builtin(__builtin_amdgcn_global_load_async_to_lds_b128)
#    define HAVE_ASYNC_LDS 1
#  endif
#endif

#ifdef HAVE_ASYNC_LDS
__device__ __forceinline__ void async_cp16(void* lds, const void* g) {
  // signature (probe-confirmed by diagnostic): (AS1 v4i*, AS3 v4i*, imm, imm)
  __builtin_amdgcn_global_load_async_to_lds_b128(
      (__attribute__((address_space(1))) v4i*)g,
      (__attribute__((address_space(3))) v4i*)lds, 0, 0);
}
__device__ __forceinline__ void async_wait0() {
#  if __has_builtin(__builtin_amdgcn_s_wait_asynccnt)
  __builtin_amdgcn_s_wait_asynccnt(0);
#  else
  asm volatile("s_wait_asynccnt 0" ::: "memory");
#  endif
}
#endif

// ---------------------------------------------------------------------------
// zero the aggregation buffer (d_ws is poisoned 0xAA; must zero every call)
// ---------------------------------------------------------------------------
__global__ void zero_f4(v4f* __restrict__ p, int n4) {
  int i = blockIdx.x * blockDim.x + threadIdx.x;
  if (i < n4) {
    v4f z = {0.f, 0.f, 0.f, 0.f};
    p[i] = z;
  }
}

// ---------------------------------------------------------------------------
// Edge scatter: one wave32 per edge.  Lane l reads float4 l of the 512B source
// row (coalesced global_load_b128) and does 4 global_atomic_add_f32 into the
// destination row.  feat/agg/indices are all L2-resident (<= 65 MB total).
// ---------------------------------------------------------------------------
__global__ void __launch_bounds__(256)
gin_scatter(const float* __restrict__ feat, const int* __restrict__ src,
            const int* __restrict__ dst, float* __restrict__ agg, int n_edges) {
  int gid  = blockIdx.x * blockDim.x + threadIdx.x;
  int edge = gid >> 5;
  int lane = gid & 31;
  if (edge >= n_edges) return;
  int s = src[edge];
  int d = dst[edge];
  v4f v = ((const v4f*)(feat + (size_t)s * DIM))[lane];
  float* ap = agg + (size_t)d * DIM + lane * 4;
  atomicAdd(ap + 0, v.x);
  atomicAdd(ap + 1, v.y);
  atomicAdd(ap + 2, v.z);
  atomicAdd(ap + 3, v.w);
}

// ---------------------------------------------------------------------------
// Fused hidden layer: out = relu((h + agg) @ W + b), 128x128 W.
// Block = 256 threads (8 waves) -> 64 x 128 output tile.
// Wave (m = wave/2, n-half = wave&1) holds four 16x16 f32 accumulators and
// issues V_WMMA_F32_16X16X4_F32 over K in steps of 4 (two 64-row W slices
// staged in LDS to keep static LDS at 64 KB).
// ---------------------------------------------------------------------------
__global__ void __launch_bounds__(256)
gin_gemm_relu(const float* __restrict__ h, const float* __restrict__ agg,
              const float* __restrict__ W, const float* __restrict__ bias,
              float* __restrict__ out) {
  __shared__ float lX[64 * DIM];   // 32 KB : X = h + agg tile
  __shared__ float lW[64 * DIM];   // 32 KB : one 64-row K-slice of W

  const int tid  = threadIdx.x;
  const int wave = tid >> 5;
  const int lane = tid & 31;
  const int half = lane >> 4;      // K-pair select for 16x4 f32 A-frag layout
  const int l15  = lane & 15;
  const int row0 = blockIdx.x * 64;

  // stage X = h + agg (64 rows x 128 cols), coalesced float4, zero-pad M tail
  #pragma unroll
  for (int i = 0; i < 8; ++i) {
    int e = tid + 256 * i;               // float4 index, 2048 total
    int r = e >> 5, c = e & 31;
    int gr = row0 + r;
    v4f x = {0.f, 0.f, 0.f, 0.f};
    if (gr < N_NODES) {
      v4f a = ((const v4f*)(h   + (size_t)gr * DIM))[c];
      v4f g = ((const v4f*)(agg + (size_t)gr * DIM))[c];
      x = a + g;
    }
    ((v4f*)lX)[e] = x;
  }

  const int mtile = wave >> 1;
  const int nb0   = (wave & 1) * 64;     // this wave's 64-column strip
  v8f acc[4] = {};                       // four 16x16 f32 accumulators

  // A-frag base: lanes 0-15 read K={k,k+1}, lanes 16-31 read K={k+2,k+3}
  const float* Xrow = lX + (mtile * 16 + l15) * DIM + half * 2;

  for (int kh = 0; kh < 2; ++kh) {
    // ---- stage W rows [kh*64, kh*64+64) into LDS ----
    const float* Wsrc = W + (size_t)kh * 64 * DIM;
#ifdef HAVE_ASYNC_LDS
    #pragma unroll
    for (int i = 0; i < 8; ++i) {
      int e = tid + 256 * i;
      async_cp16((v4f*)lW + e, (const v4f*)Wsrc + e);
    }
    async_wait0();
#else
    #pragma unroll
    for (int i = 0; i < 8; ++i) {
      int e = tid + 256 * i;
      ((v4f*)lW)[e] = ((const v4f*)Wsrc)[e];
    }
#endif
    __syncthreads();                     // X (first iter) + W slice visible

    const float* Xk = Xrow + kh * 64;
    #pragma unroll 4
    for (int k0 = 0; k0 < 64; k0 += 4) {
      v2f a = *(const v2f*)(Xk + k0);    // ds_load_b64 A-frag
      const float* Bp = lW + (k0 + half * 2) * DIM + nb0 + l15;
      #pragma unroll
      for (int t = 0; t < 4; ++t) {
        v2f b;
        b.x = Bp[t * 16];                // B-frag rows K, K+1 of n-tile t
        b.y = Bp[t * 16 + DIM];
        acc[t] = __builtin_amdgcn_wmma_f32_16x16x4_f32(
            false, a, false, b, (short)0, acc[t], false, false);
      }
    }
    __syncthreads();                     // compute done before re-staging lW
  }

  // bias + ReLU + store (C/D layout: VGPR j -> M = j + half*8, N = l15)
  #pragma unroll
  for (int t = 0; t < 4; ++t) {
    int col = nb0 + t * 16 + l15;
    float bv = bias[col];
    #pragma unroll
    for (int j = 0; j < 8; ++j) {
      int gr = row0 + mtile * 16 + half * 8 + j;
      if (gr < N_NODES) {
        float v = acc[t][j] + bv;
        out[(size_t)gr * DIM + col] = fmaxf(v, 0.f);
      }
    }
  }
}

// ---------------------------------------------------------------------------
// Final layer: out = (h + agg) @ W3 + b3, W3 is 128x4.  W3 zero-padded to a
// 128x16 LDS tile; one 16x16 WMMA accumulator per wave, 8 waves -> 128 rows.
// Only columns 0..3 are stored.
// ---------------------------------------------------------------------------
__global__ void __launch_bounds__(256)
gin_gemm_final(const float* __restrict__ h, const float* __restrict__ agg,
               const float* __restrict__ W3, const float* __restrict__ b3,
               float* __restrict__ out) {
  __shared__ float lW[DIM * 16];         // 8 KB, cols 4..15 zero

  const int tid  = threadIdx.x;
  const int wave = tid >> 5;
  const int lane = tid & 31;
  const int half = lane >> 4;
  const int l15  = lane & 15;
  const int row0 = blockIdx.x * 128;

  for (int i = tid; i < DIM * 16; i += 256) {
    int k = i >> 4, n = i & 15;
    lW[i] = (n < COUT) ? W3[k * COUT + n] : 0.f;
  }
  __syncthreads();

  const int grow = row0 + wave * 16 + l15;
  const bool rok = grow < N_NODES;
  const float* hp = h   + (size_t)(rok ? grow : 0) * DIM + half * 2;
  const float* gp = agg + (size_t)(rok ? grow : 0) * DIM + half * 2;

  v8f acc = {};
  #pragma unroll 4
  for (int k0 = 0; k0 < DIM; k0 += 4) {
    v2f a = {0.f, 0.f};
    if (rok) a = *(const v2f*)(hp + k0) + *(const v2f*)(gp + k0);
    v2f b;
    b.x = lW[(k0 + half * 2) * 16 + l15];
    b.y = lW[(k0 + half * 2 + 1) * 16 + l15];
    acc = __builtin_amdgcn_wmma_f32_16x16x4_f32(
        false, a, false, b, (short)0, acc, false, false);
  }

  if (l15 < COUT) {
    float bv = b3[l15];
    #pragma unroll
    for (int j = 0; j < 8; ++j) {
      int gr = row0 + wave * 16 + half * 8 + j;
      if (gr < N_NODES) out[(size_t)gr * COUT + l15] = acc[j] + bv;
    }
  }
}

// ---------------------------------------------------------------------------
extern "C" void kernel_launch(void* const* d_in, const int* in_sizes, int n_in,
                              void* d_out, int out_size, void* d_ws, size_t ws_size,
                              hipStream_t stream) {
  const float* feat = (const float*)d_in[0];
  const int*   src  = (const int*)d_in[1];
  const int*   dst  = (const int*)d_in[2];
  const float* W0 = (const float*)d_in[3];  const float* b0 = (const float*)d_in[4];
  const float* W1 = (const float*)d_in[5];  const float* b1 = (const float*)d_in[6];
  const float* W2 = (const float*)d_in[7];  const float* b2 = (const float*)d_in[8];
  const float* W3 = (const float*)d_in[9];  const float* b3 = (const float*)d_in[10];
  const int E = in_sizes[1];

  const size_t fsz = (size_t)N_NODES * DIM;     // 6.4M floats = 25.6 MB
  float* agg = (float*)d_ws;
  float* hA  = agg + fsz;
  float* hB  = hA + fsz;

  const dim3 blk(256);
  const int n4    = (int)(fsz / 4);
  const int zgrid = (n4 + 255) / 256;
  const int sgrid = (int)(((size_t)E * 32 + 255) / 256);
  const int ggrid = (N_NODES + 63) / 64;
  const int fgrid = (N_NODES + 127) / 128;

  // layer 0: feat -> hA
  zero_f4<<<zgrid, blk, 0, stream>>>((v4f*)agg, n4);
  gin_scatter<<<sgrid, blk, 0, stream>>>(feat, src, dst, agg, E);
  gin_gemm_relu<<<ggrid, blk, 0, stream>>>(feat, agg, W0, b0, hA);

  // layer 1: hA -> hB
  zero_f4<<<zgrid, blk, 0, stream>>>((v4f*)agg, n4);
  gin_scatter<<<sgrid, blk, 0, stream>>>(hA, src, dst, agg, E);
  gin_gemm_relu<<<ggrid, blk, 0, stream>>>(hA, agg, W1, b1, hB);

  // layer 2: hB -> hA
  zero_f4<<<zgrid, blk, 0, stream>>>((v4f*)agg, n4);
  gin_scatter<<<sgrid, blk, 0, stream>>>(hB, src, dst, agg, E);
  gin_gemm_relu<<<ggrid, blk, 0, stream>>>(hB, agg, W2, b2, hA);

  // layer 3 (final, no relu): hA -> d_out (50000 x 4)
  zero_f4<<<zgrid, blk, 0, stream>>>((v4f*)agg, n4);
  gin_scatter<<<sgrid, blk, 0, stream>>>(hA, src, dst, agg, E);
  gin_gemm_final<<<fgrid, blk, 0, stream>>>(hA, agg, W3, b3, (float*)d_out);
}